// UpCaps_13632226197856
// MI455X (gfx1250) — compile-verified
//
#include <hip/hip_runtime.h>

// ---------------------------------------------------------------------------
// MI455X plan:
//  Kernel 1 (voting): stride-2 transposed conv as implicit GEMM on the
//    matrix engine with V_WMMA_F32_16X16X4_F32 (fp32-exact).
//    Block = one output-parity class (p,q): all 8 waves share the tap set.
//    Each tap's 16x128 weight slab is staged once into LDS in K-pair-
//    interleaved float2 form, so a B fragment is a single ds_load_b64 into
//    an even-aligned VGPR pair (direct WMMA operand). All 8 fragments of a
//    K-group are staged into distinct pairs first so the ds loads clause up
//    and the 8 WMMAs retire behind staggered s_wait_dscnt (latency hidden).
//    Row pitch 144 float2 -> the two half-waves hit disjoint 32-bank groups.
//    A fragments: unconditional clamped global loads + v_cndmask halo mask.
//  Kernel 2 (routing): per-pixel independent -> one streaming pass over the
//    151MB votes buffer laid out [h][w][b][n] (4KB/pixel contiguous),
//    3 routing iterations fully in registers + LDS.
// ---------------------------------------------------------------------------

typedef __attribute__((ext_vector_type(2))) float v2f;
typedef __attribute__((ext_vector_type(8))) float v8f;

#define H_IN   96
#define W_IN   96
#define HW_IN  (H_IN * W_IN)
#define CIN    16
#define B_     8
#define A_     8
#define COUT   16
#define N_     128            // A_*COUT
#define H_OUT  192
#define W_OUT  192
#define NPIX   (H_OUT * W_OUT)
#define WP2    144            // float2 pitch per K-pair row (bank-safe)

// ======================= voting: WMMA implicit GEMM ========================
// votes workspace layout: [oh][ow][b][n], n = a*16 + co
__global__ void __launch_bounds__(256)
vote_wmma_kernel(const float* __restrict__ x,
                 const float* __restrict__ wt,
                 float* __restrict__ votes)
{
    // 8 K-pair rows x 144 float2 = 9216 B: row k holds {w[2k][n], w[2k+1][n]}
    __shared__ v2f wS[8 * WP2];

    const int tid   = threadIdx.x;
    const int lane  = tid & 31;
    const int laneM = lane & 15;   // M index (A frag) / N index (B,C frags)
    const int laneH = lane >> 4;   // K-half selector
    const int wave  = tid >> 5;

    // parity class is per-block -> uniform tap loop for all 8 waves
    const int cls = blockIdx.x / 576;       // 0..3
    const int rem = blockIdx.x % 576;
    const int p   = cls >> 1;               // oh parity
    const int q   = cls & 1;                // ow parity

    int t = rem * 8 + wave;                 // 0..4607: (b, oy, ox-tile)
    const int ox0 = (t % 6) * 16;  t /= 6;
    const int oy  = t % 96;        t /= 96;
    const int b   = t;                      // 0..7
    const int oh  = 2 * oy + p;

    v8f acc[8] = {};                        // 16x128 fp32 tile

    const int nTh = 3 - p;   // taps: p==0 -> kh{0,2,4}, p==1 -> kh{1,3}
    const int nTw = 3 - q;

    for (int i = 0; i < nTh; ++i) {
        const int  kh  = 2 * i + p;
        const int  ih  = oy + i - 1 + p;
        const bool okh = (unsigned)ih < H_IN;
        for (int j = 0; j < nTw; ++j) {
            const int kw = 2 * j + q;

            // ---- stage tap slab into LDS, K-pair interleaved ----
            __syncthreads();
            {
                const float4* src =
                    (const float4*)(wt + (size_t)(kh * 5 + kw) * CIN * N_);
                #pragma unroll
                for (int e = 0; e < 2; ++e) {
                    const int idx4 = tid * 2 + e;       // 0..511
                    const int idx  = idx4 * 4;          // element index
                    const int ci   = idx >> 7;          // 0..15
                    const int n0   = idx & 127;
                    const int k    = ci >> 1;           // K-pair row
                    const int sub  = ci & 1;            // component in pair
                    const float4 w4 = src[idx4];
                    ((float*)&wS[k * WP2 + n0 + 0])[sub] = w4.x;
                    ((float*)&wS[k * WP2 + n0 + 1])[sub] = w4.y;
                    ((float*)&wS[k * WP2 + n0 + 2])[sub] = w4.z;
                    ((float*)&wS[k * WP2 + n0 + 3])[sub] = w4.w;
                }
            }
            __syncthreads();

            // ---- A halo handling: clamped address + value select ----
            const int  iw  = ox0 + laneM + j - 1 + q;
            const bool ok  = okh && ((unsigned)iw < W_IN);
            const int  abase = (b * CIN * H_IN + (okh ? ih : 0)) * W_IN
                             + (((unsigned)iw < W_IN) ? iw : 0);

            #pragma unroll
            for (int kg = 0; kg < 4; ++kg) {
                // ISA 16x4 f32 A layout: lane L -> (M=L%16, K=2*(L/16)+{0,1})
                const int ci = kg * 4 + 2 * laneH;
                const float v0 = x[abase + (ci    ) * HW_IN];
                const float v1 = x[abase + (ci + 1) * HW_IN];
                v2f afrag;
                afrag.x = ok ? v0 : 0.f;
                afrag.y = ok ? v1 : 0.f;

                // B frags: 8 independent ds_load_b64 destinations so the
                // loads clause and WMMAs retire behind staggered waits
                const int brow = (kg * 2 + laneH) * WP2 + laneM;
                v2f bf[8];
                #pragma unroll
                for (int nt = 0; nt < 8; ++nt)
                    bf[nt] = wS[brow + nt * 16];
                #pragma unroll
                for (int nt = 0; nt < 8; ++nt)
                    acc[nt] = __builtin_amdgcn_wmma_f32_16x16x4_f32(
                        false, afrag, false, bf[nt],
                        (short)0, acc[nt], false, false);
            }
        }
    }

    // C layout: VGPR r, lanes 0-15 -> (M=r, N=lane), lanes 16-31 -> (M=r+8)
    #pragma unroll
    for (int nt = 0; nt < 8; ++nt) {
        #pragma unroll
        for (int r = 0; r < 8; ++r) {
            const int m  = r + 8 * laneH;
            const int ow = 2 * (ox0 + m) + q;
            votes[((size_t)(oh * W_OUT + ow) * B_ + b) * N_ + nt * 16 + laneM]
                = acc[nt][r];
        }
    }
}

// ======================= routing: per-pixel, streaming =====================
// 256 threads = 4 pixels x 64 threads; thread (b,a) holds votes[b,a,:] (16 f)
__global__ void __launch_bounds__(256)
routing_kernel(const float* __restrict__ votes,
               const int* __restrict__ nroutes_p,
               float* __restrict__ out)
{
    __shared__ float sL[4][64];        // logits staging (softmax over b)
    __shared__ float sS[4][64][17];    // preds partials (+pad vs bank conflicts)
    __shared__ float sF[4][16];        // squash factors per c

    const int tid = threadIdx.x;
    const int pp  = tid >> 6;          // pixel within block
    const int t64 = tid & 63;
    const int b   = t64 >> 3;
    const int a   = t64 & 7;
    const int pix = blockIdx.x * 4 + pp;

    const int nroutes = *nroutes_p;

    // coalesced: 4KB contiguous per pixel, 16 contiguous floats per thread
    float v[16];
    const float4* vp =
        (const float4*)(votes + ((size_t)pix * B_ + b) * N_ + a * 16);
    #pragma unroll
    for (int i = 0; i < 4; ++i) {
        float4 t = vp[i];
        v[4*i+0] = t.x; v[4*i+1] = t.y; v[4*i+2] = t.z; v[4*i+3] = t.w;
    }

    float logit = 0.f;

    for (int r = 0; r < nroutes; ++r) {
        // --- softmax over batch b at fixed a ---
        sL[pp][t64] = logit;
        __syncthreads();
        float mx = -3.402823466e38f;
        #pragma unroll
        for (int bb = 0; bb < 8; ++bb) mx = fmaxf(mx, sL[pp][bb * 8 + a]);
        float den = 0.f;
        #pragma unroll
        for (int bb = 0; bb < 8; ++bb) den += expf(sL[pp][bb * 8 + a] - mx);
        const float ls = expf(logit - mx) / den;

        // --- preds partial: v[c] * ls, reduce over b (tree in LDS) ---
        #pragma unroll
        for (int c = 0; c < 16; ++c) sS[pp][t64][c] = v[c] * ls;
        __syncthreads();
        #pragma unroll
        for (int off = 4; off >= 1; off >>= 1) {
            if (b < off) {
                #pragma unroll
                for (int c = 0; c < 16; ++c)
                    sS[pp][b * 8 + a][c] += sS[pp][(b + off) * 8 + a][c];
            }
            __syncthreads();
        }

        // --- squash over atoms a: factor(c) = s/(1+s^2), s = ||preds(:,c)|| ---
        if (t64 < 16) {
            float ss = 0.f;
            #pragma unroll
            for (int aa = 0; aa < 8; ++aa) {
                const float pv = sS[pp][aa][t64];
                ss += pv * pv;
            }
            sF[pp][t64] = sqrtf(ss) / (1.f + ss);
        }
        __syncthreads();

        if (r + 1 < nroutes) {
            // logits = softmax(logits) + sum_c votes*preds_squashed
            float upd = 0.f;
            #pragma unroll
            for (int c = 0; c < 16; ++c)
                upd += v[c] * sS[pp][a][c] * sF[pp][c];
            logit = ls + upd;
        } else {
            // final squashed preds -> out[(a*16+c)*NPIX + pix]
            #pragma unroll
            for (int k = 0; k < 2; ++k) {
                const int n  = t64 + 64 * k;
                const int ao = n >> 4, co = n & 15;
                out[(size_t)n * NPIX + pix] = sS[pp][ao][co] * sF[pp][co];
            }
        }
    }
}

// ============================ launcher =====================================
extern "C" void kernel_launch(void* const* d_in, const int* in_sizes, int n_in,
                              void* d_out, int out_size, void* d_ws, size_t ws_size,
                              hipStream_t stream) {
    (void)in_sizes; (void)n_in; (void)out_size; (void)ws_size;
    const float* x  = (const float*)d_in[0];   // (8,16,96,96)
    const float* wt = (const float*)d_in[1];   // (5,5,16,8,16)
    const int*   nr = (const int*)d_in[2];     // num_routes
    float* votes = (float*)d_ws;               // [192][192][8][128] = 151 MB
    float* out   = (float*)d_out;              // (8,16,192,192)

    // 4 parity classes x 576 blocks; each block = 8 wave-tiles of one class
    vote_wmma_kernel<<<2304, 256, 0, stream>>>(x, wt, votes);
    // 36864 pixels / 4 per block
    routing_kernel<<<9216, 256, 0, stream>>>(votes, nr, out);
}